// NMI_Loss_17566416241189
// MI455X (gfx1250) — compile-verified
//
#include <hip/hip_runtime.h>
#include <hip/hip_bf16.h>

#define NBINS 64
#define NB2   4096   // 64*64
#define TPB   256

typedef float v4f __attribute__((ext_vector_type(4)));
typedef __attribute__((ext_vector_type(2))) float v2f;
typedef __attribute__((ext_vector_type(8))) float v8f;

// ---------------------------------------------------------------------------
// Kernel 1: streaming joint histogram. Per-block LDS hist (ds_add_u32), then
// either a clean partial dump (partial mode) or global atomics (fallback).
// ---------------------------------------------------------------------------
__global__ __launch_bounds__(TPB) void nmi_hist(const v4f* __restrict__ x4,
                                                const v4f* __restrict__ y4,
                                                const float* __restrict__ x,
                                                const float* __restrict__ y,
                                                unsigned* __restrict__ outp,
                                                int n4, int n, int use_atomic) {
  __shared__ unsigned lh[NB2];
  for (int i = threadIdx.x; i < NB2; i += TPB) lh[i] = 0u;
  __syncthreads();

  const int stride = gridDim.x * TPB;
  for (int i = blockIdx.x * TPB + threadIdx.x; i < n4; i += stride) {
    v4f a = __builtin_nontemporal_load(x4 + i);
    v4f b = __builtin_nontemporal_load(y4 + i);
#pragma unroll
    for (int j = 0; j < 4; ++j) {
      float v = a[j] + b[j];                       // x*MAX_VAL + y, MAX_VAL=1
      if (v >= 0.0f && v <= 1.0f) {                // histc: out-of-range dropped
        int idx = (int)(v * 4096.0f);              // width = 2^-12 exact
        idx = idx > (NB2 - 1) ? (NB2 - 1) : idx;   // v==1 -> last bin
        atomicAdd(&lh[idx], 1u);                   // ds_add_u32
      }
    }
  }
  // tail (n % 4), generically handled by one thread
  if (blockIdx.x == 0 && threadIdx.x == 0) {
    for (int i = n4 * 4; i < n; ++i) {
      float v = x[i] + y[i];
      if (v >= 0.0f && v <= 1.0f) {
        int idx = (int)(v * 4096.0f);
        idx = idx > (NB2 - 1) ? (NB2 - 1) : idx;
        atomicAdd(&lh[idx], 1u);
      }
    }
  }
  __syncthreads();

  if (use_atomic) {
    for (int i = threadIdx.x; i < NB2; i += TPB) {
      unsigned c = lh[i];
      if (c) atomicAdd(outp + i, c);               // global_atomic_add_u32
    }
  } else {
    unsigned* dst = outp + (size_t)blockIdx.x * NB2;
    for (int i = threadIdx.x; i < NB2; i += TPB) dst[i] = lh[i];
  }
}

// ---------------------------------------------------------------------------
// Kernel 1b (fallback path only): zero final hist.
// ---------------------------------------------------------------------------
__global__ void nmi_zero(unsigned* __restrict__ h) {
  h[blockIdx.x * blockDim.x + threadIdx.x] = 0u;
}

// ---------------------------------------------------------------------------
// Kernel 2: reduce per-block partial histograms into final 4096-bin hist.
// ---------------------------------------------------------------------------
__global__ __launch_bounds__(TPB) void nmi_reduce(const unsigned* __restrict__ part,
                                                  unsigned* __restrict__ hist,
                                                  int nparts) {
  int bin = blockIdx.x * TPB + threadIdx.x;
  if (bin < NB2) {
    unsigned s = 0u;
    for (int p = 0; p < nparts; ++p) s += part[(size_t)p * NB2 + bin];
    hist[bin] = s;
  }
}

// ---------------------------------------------------------------------------
// WMMA marginal sums of a 64x64 f32 matrix in LDS using v_wmma_f32_16x16x4_f32.
// Row sums: C += A(16x4 block of M) * ones(4x16); col sums: feed M^T into A.
// Only the documented A and C/D layouts are relied upon (B is all-ones).
// Executed by one full wave (lanes 0..31, EXEC all ones).
// ---------------------------------------------------------------------------
__device__ inline void wmma_sums64(const float* __restrict__ m,
                                   float* __restrict__ rs, int transpose) {
  const int lane = threadIdx.x & 31;
  const int half = lane >> 4;     // 0: K={0,1}, 1: K={2,3} per A layout
  const int r    = lane & 15;     // M within 16-row block
  v2f ones; ones.x = 1.0f; ones.y = 1.0f;
  for (int rb = 0; rb < 4; ++rb) {
    v8f c = {0.f, 0.f, 0.f, 0.f, 0.f, 0.f, 0.f, 0.f};
    const int row = rb * 16 + r;
    for (int kb = 0; kb < 16; ++kb) {
      const int k0 = kb * 4 + half * 2;
      v2f a;
      if (!transpose) {           // A[m][k] = M[row][k]   -> row sums
        a.x = m[row * 64 + k0];
        a.y = m[row * 64 + k0 + 1];
      } else {                    // A[m][k] = M[k][row]   -> column sums
        a.x = m[k0 * 64 + row];
        a.y = m[(k0 + 1) * 64 + row];
      }
      c = __builtin_amdgcn_wmma_f32_16x16x4_f32(false, a, false, ones,
                                                (short)0, c, false, false);
    }
    // C layout: VGPR v, lanes 0-15 -> (M=v, N=lane); lanes 16-31 -> (M=v+8).
    // Every column equals the sums; extract column N==0 via lanes 0 and 16.
    if (r == 0) {
#pragma unroll
      for (int v = 0; v < 8; ++v) rs[rb * 16 + half * 8 + v] = c[v];
    }
  }
}

__device__ inline float block_reduce(float v, float* red) {
  int tid = threadIdx.x;
  red[tid] = v;
  __syncthreads();
  for (int s = TPB / 2; s > 0; s >>= 1) {
    if (tid < s) red[tid] += red[tid + s];
    __syncthreads();
  }
  float r = red[0];
  __syncthreads();
  return r;
}

// ---------------------------------------------------------------------------
// Kernel 3: finalize — MI + entropies -> -NMI scalar. Single block.
// ---------------------------------------------------------------------------
__global__ __launch_bounds__(TPB) void nmi_finalize(const unsigned* __restrict__ g_hist,
                                                    float* __restrict__ out) {
  __shared__ float histf[NB2];
  __shared__ float pxyf[NB2];
  __shared__ float h1[NBINS], h2[NBINS], pxs[NBINS], pys[NBINS];
  __shared__ float red[TPB];

  const int tid = threadIdx.x;

  float acc = 0.0f;
  for (int i = tid; i < NB2; i += TPB) {
    float v = (float)g_hist[i];
    histf[i] = v;
    acc += v;
  }
  __syncthreads();
  const float total = block_reduce(acc, red);      // exact integer (< 2^24)

  for (int i = tid; i < NB2; i += TPB) pxyf[i] = histf[i] / total;
  __syncthreads();

  if (tid < 32) {                                  // one wave, EXEC all-ones
    wmma_sums64(histf, h1, 0);                     // row sums of counts
    wmma_sums64(histf, h2, 1);                     // col sums of counts
    wmma_sums64(pxyf, pxs, 0);                     // px = sum(pxy, axis=1)
    wmma_sums64(pxyf, pys, 1);                     // py = sum(pxy, axis=0)
  }
  __syncthreads();

  // mutual information — includes the torch quirk: zero bins contribute
  // 1*log2(1/(px*py+eps)) because pxy_safe replaces zeros with ones.
  acc = 0.0f;
  for (int i = tid; i < NB2; i += TPB) {
    float p  = pxyf[i];
    float ps = (p != 0.0f) ? p : 1.0f;
    float pp = pxs[i >> 6] * pys[i & 63];
    acc += ps * log2f(ps / (pp + 1e-6f));
  }
  const float mi = block_reduce(acc, red);

  // marginal "entropies" on raw counts: H = -sum(hs*log2(hs)), hs = h or 1
  float e = 0.0f;
  if (tid < NBINS) {
    float a = h1[tid];
    if (a != 0.0f) e += a * log2f(a);
    float b = h2[tid];
    if (b != 0.0f) e += b * log2f(b);
  }
  const float esum = block_reduce(e, red);         // S1 + S2

  if (tid == 0) {
    float ent = -esum;                             // H1 + H2 (negative)
    float nmi = 2.0f * mi / (ent + 1e-6f);
    out[0] = -nmi;
  }
}

// ---------------------------------------------------------------------------
extern "C" void kernel_launch(void* const* d_in, const int* in_sizes, int n_in,
                              void* d_out, int out_size, void* d_ws, size_t ws_size,
                              hipStream_t stream) {
  const float* x = (const float*)d_in[0];
  const float* y = (const float*)d_in[1];
  const int n  = in_sizes[0];
  const int n4 = n / 4;

  unsigned* hist = (unsigned*)d_ws;                // first 16 KB: final hist
  const size_t HB = (size_t)NB2 * sizeof(unsigned);

  int blocks = 512;                                // memory-parallelism target
  int use_atomic = 1;
  unsigned* partials = hist + NB2;
  if (ws_size >= HB) {
    size_t avail = (ws_size - HB) / HB;            // partial hists that fit
    if (avail >= 32) {
      use_atomic = 0;
      if ((size_t)blocks > avail) blocks = (int)avail;
    }
  }

  if (use_atomic) {
    nmi_zero<<<NB2 / TPB, TPB, 0, stream>>>(hist);
    nmi_hist<<<blocks, TPB, 0, stream>>>((const v4f*)x, (const v4f*)y, x, y,
                                         hist, n4, n, 1);
  } else {
    nmi_hist<<<blocks, TPB, 0, stream>>>((const v4f*)x, (const v4f*)y, x, y,
                                         partials, n4, n, 0);
    nmi_reduce<<<NB2 / TPB, TPB, 0, stream>>>(partials, hist, blocks);
  }
  nmi_finalize<<<1, TPB, 0, stream>>>(hist, (float*)d_out);
}